// sgcGCN_16286515986688
// MI455X (gfx1250) — compile-verified
//
#include <hip/hip_runtime.h>

typedef float v2f __attribute__((ext_vector_type(2)));
typedef float v8f __attribute__((ext_vector_type(8)));

#define THREADS 256

// ---------------- degree / normalization ----------------

__global__ void k_init_deg(float* __restrict__ deg, int n) {
  int i = blockIdx.x * blockDim.x + threadIdx.x;
  if (i < n) deg[i] = 1.0f;  // self-loop contribution
}

__global__ void k_scatter_deg(const int* __restrict__ dst, float* __restrict__ deg, int E) {
  int e = blockIdx.x * blockDim.x + threadIdx.x;
  if (e < E) atomicAdd(&deg[dst[e]], 1.0f);
}

__global__ void k_dis(const float* __restrict__ deg, float* __restrict__ dis, int n) {
  int i = blockIdx.x * blockDim.x + threadIdx.x;
  if (i < n) {
    float d = deg[i];
    dis[i] = (d > 0.0f) ? (1.0f / sqrtf(d)) : 0.0f;
  }
}

// ---------------- propagation: h <- S h ----------------
// Self-loop (norm = dis[i]^2) folded into accumulator init.
__global__ void k_prop_init(const float* __restrict__ in, const float* __restrict__ dis,
                            float* __restrict__ out, int n, int C) {
  int idx = blockIdx.x * blockDim.x + threadIdx.x;
  int total = n * C;
  if (idx < total) {
    int i = idx / C;
    float w = dis[i];
    out[idx] = w * w * in[idx];
  }
}

// Edge scatter: out[dst] += dis[src]*dis[dst] * in[src]  (per channel)
__global__ void k_prop_edges(const int* __restrict__ src, const int* __restrict__ dst,
                             const float* __restrict__ dis, const float* __restrict__ in,
                             float* __restrict__ out, int E, int C) {
  int idx = blockIdx.x * blockDim.x + threadIdx.x;
  int total = E * C;
  if (idx < total) {
    int e = idx / C;
    int c = idx - e * C;
    int s = src[e];
    int d = dst[e];
    float w = dis[s] * dis[d];
    atomicAdd(&out[d * C + c], w * in[s * C + c]);
  }
}

// ---------------- fp32 WMMA GEMM: D = act(A[MxK] @ B[KxN] + bias) ----------------
// Block = 256 threads = 8 waves; each wave owns one 16x16 output tile.
// rowTiles * (N/16) must equal 8.  K must be a multiple of 4.
__global__ void k_wmma_gemm(const float* __restrict__ A, const float* __restrict__ B,
                            const float* __restrict__ bias, float* __restrict__ D,
                            int M, int K, int N, int rowTiles, int doRelu) {
  const int wave = threadIdx.x >> 5;
  const int lane = threadIdx.x & 31;
  const int half = lane >> 4;   // K sub-pair select: lanes 0-15 -> K{0,1}, 16-31 -> K{2,3}
  const int lid  = lane & 15;
  const int rowTile = wave % rowTiles;
  const int colTile = wave / rowTiles;
  const int m0 = (blockIdx.x * rowTiles + rowTile) << 4;
  const int n0 = colTile << 4;

  // A operand row (clamped so OOB tiles read in-bounds; their stores are guarded)
  const int mA = m0 + lid;
  const int mAc = (mA < M) ? mA : (M - 1);
  const int nB = n0 + lid;
  const float* __restrict__ Arow = A + (long)mAc * K;

  v8f acc = {};
  for (int k = 0; k < K; k += 4) {
    const int ka = k + half * 2;
    v2f a, b;
    a.x = Arow[ka];
    a.y = Arow[ka + 1];
    b.x = B[ka * N + nB];
    b.y = B[(ka + 1) * N + nB];
    // (neg_a, A, neg_b, B, c_mod, C, reuse_a, reuse_b)
    acc = __builtin_amdgcn_wmma_f32_16x16x4_f32(false, a, false, b, (short)0, acc,
                                                false, false);
  }

  const float bv = bias[nB];
  const int mBase = m0 + half * 8;  // C/D layout: VGPR r -> M = m0 + half*8 + r, N = n0+lid
#pragma unroll
  for (int r = 0; r < 8; ++r) {
    int m = mBase + r;
    if (m < M) {
      float v = acc[r] + bv;
      if (doRelu) v = fmaxf(v, 0.0f);
      D[(long)m * N + nB] = v;
    }
  }
}

// ---------------- row-wise log_softmax (in place) ----------------
__global__ void k_log_softmax(float* __restrict__ out, int M, int N) {
  int row = blockIdx.x * blockDim.x + threadIdx.x;
  if (row < M) {
    float* p = out + (long)row * N;
    float mx = p[0];
    for (int c = 1; c < N; ++c) mx = fmaxf(mx, p[c]);
    float s = 0.0f;
    for (int c = 0; c < N; ++c) s += expf(p[c] - mx);
    float ls = logf(s);
    for (int c = 0; c < N; ++c) p[c] = p[c] - mx - ls;
  }
}

static inline int cdiv(int a, int b) { return (a + b - 1) / b; }

extern "C" void kernel_launch(void* const* d_in, const int* in_sizes, int n_in,
                              void* d_out, int out_size, void* d_ws, size_t ws_size,
                              hipStream_t stream) {
  const float* x  = (const float*)d_in[0];
  const int*   ei = (const int*)d_in[1];
  const float* W1 = (const float*)d_in[2];
  const float* b1 = (const float*)d_in[3];
  const float* W2 = (const float*)d_in[4];
  const float* b2 = (const float*)d_in[5];
  float* out = (float*)d_out;

  const int HID  = in_sizes[3];            // 64
  const int OUTC = in_sizes[5];            // 16
  const int INC  = in_sizes[2] / HID;      // 100
  const int Nn   = in_sizes[0] / INC;      // 50000
  const int E    = in_sizes[1] / 2;        // 800000
  const int* src = ei;        // edge_index[0] : message sources
  const int* dst = ei + E;    // edge_index[1] : aggregation targets

  // workspace layout (floats): deg | dis | bufA (Nn*INC) | bufB (Nn*INC)
  float* ws   = (float*)d_ws;
  float* deg  = ws;
  float* dis  = deg + Nn;
  float* bufA = dis + Nn;
  float* bufB = bufA + (size_t)Nn * INC;

  // --- gcn_norm ---
  k_init_deg<<<cdiv(Nn, THREADS), THREADS, 0, stream>>>(deg, Nn);
  k_scatter_deg<<<cdiv(E, THREADS), THREADS, 0, stream>>>(dst, deg, E);
  k_dis<<<cdiv(Nn, THREADS), THREADS, 0, stream>>>(deg, dis, Nn);

  // --- layer 1: S^2 x ---
  {
    int tN = Nn * INC, tE = E * INC;
    k_prop_init <<<cdiv(tN, THREADS), THREADS, 0, stream>>>(x, dis, bufA, Nn, INC);
    k_prop_edges<<<cdiv(tE, THREADS), THREADS, 0, stream>>>(src, dst, dis, x, bufA, E, INC);
    k_prop_init <<<cdiv(tN, THREADS), THREADS, 0, stream>>>(bufA, dis, bufB, Nn, INC);
    k_prop_edges<<<cdiv(tE, THREADS), THREADS, 0, stream>>>(src, dst, dis, bufA, bufB, E, INC);
  }
  // hid = relu(bufB @ W1 + b1) -> bufA   (N=64 -> 4 col tiles, 2 row tiles/block)
  {
    int rowTiles = 8 / (HID / 16);  // = 2
    k_wmma_gemm<<<cdiv(Nn, rowTiles * 16), THREADS, 0, stream>>>(
        bufB, W1, b1, bufA, Nn, INC, HID, rowTiles, /*relu=*/1);
  }

  // --- layer 2: S^2 hid ---
  {
    int tN = Nn * HID, tE = E * HID;
    k_prop_init <<<cdiv(tN, THREADS), THREADS, 0, stream>>>(bufA, dis, bufB, Nn, HID);
    k_prop_edges<<<cdiv(tE, THREADS), THREADS, 0, stream>>>(src, dst, dis, bufA, bufB, E, HID);
    k_prop_init <<<cdiv(tN, THREADS), THREADS, 0, stream>>>(bufB, dis, bufA, Nn, HID);
    k_prop_edges<<<cdiv(tE, THREADS), THREADS, 0, stream>>>(src, dst, dis, bufB, bufA, E, HID);
  }
  // out = bufA @ W2 + b2   (N=16 -> 1 col tile, 8 row tiles/block)
  {
    int rowTiles = 8 / (OUTC / 16);  // = 8
    k_wmma_gemm<<<cdiv(Nn, rowTiles * 16), THREADS, 0, stream>>>(
        bufA, W2, b2, out, Nn, HID, OUTC, rowTiles, /*relu=*/0);
  }

  // --- log_softmax rows ---
  k_log_softmax<<<cdiv(Nn, THREADS), THREADS, 0, stream>>>(out, Nn, OUTC);
}